// MHELoss_87033217286635
// MI455X (gfx1250) — compile-verified
//
#include <hip/hip_runtime.h>
#include <hip/hip_bf16.h>

#define DCOLS 128
#define EPSF  1e-12f

typedef __attribute__((ext_vector_type(2))) float v2f;
typedef __attribute__((ext_vector_type(8))) float v8f;

// ds_swizzle_b32 group-of-32 xor-butterfly: offset = {0, xor[14:10], or=0, and=0x1F}
// (ISA ch.11 swizzle: SWAPX1=0x041F, X2=0x081F, X4=0x101F, X8=0x201F, X16=0x401F)
#define SWZ_ADD(x, OFF) \
    ((x) + __int_as_float(__builtin_amdgcn_ds_swizzle(__float_as_int(x), (OFF))))

__device__ __forceinline__ float wave_reduce_add(float x) {
    x = SWZ_ADD(x, 0x041F);   // xor 1
    x = SWZ_ADD(x, 0x081F);   // xor 2
    x = SWZ_ADD(x, 0x101F);   // xor 4
    x = SWZ_ADD(x, 0x201F);   // xor 8
    x = SWZ_ADD(x, 0x401F);   // xor 16
    return x;
}

// ---------------------------------------------------------------------------
// K0: zero workspace scalars (harness poisons ws with 0xAA before timing)
// ws layout: [0..127] = s column sums, [128] = ssq, [129] = total
// ---------------------------------------------------------------------------
__global__ void k0_zero(float* ws, int n) {
    int i = blockIdx.x * blockDim.x + threadIdx.x;
    if (i < n) ws[i] = 0.0f;
}

// ---------------------------------------------------------------------------
// K1: streaming pass over weight (102.4 MB @ 23.3 TB/s ~= 4.4us, HBM-bound).
// One wave per row, 2 rows per iteration: lane t loads cols 4t..4t+3 as b128
// (512B coalesced/row); both loads issue before either reduction chain, and
// the two 5-step ds_swizzle butterflies interleave for ILP. Column sums stay
// per-lane (zero cross-lane traffic), block-combined in LDS, few global
// atomics. Default RT hints keep weight resident in the 192MB L2 for K2.
// ---------------------------------------------------------------------------
__global__ __launch_bounds__(256) void k1_stream(
        const float* __restrict__ W, int C,
        float* __restrict__ g_s, float* __restrict__ g_ssq) {
    __shared__ float s_block[DCOLS];
    __shared__ float ssq_block;

    int tid = threadIdx.x;
    if (tid < DCOLS) s_block[tid] = 0.0f;
    if (tid == 0)    ssq_block = 0.0f;
    __syncthreads();

    int lane  = tid & 31;
    int wave  = tid >> 5;
    int wpb   = blockDim.x >> 5;
    int gwave = blockIdx.x * wpb + wave;
    int nwav  = gridDim.x * wpb;

    float4 s_acc = make_float4(0.f, 0.f, 0.f, 0.f);
    float  ssq_acc = 0.f;

    for (int r = gwave; r < C; r += 2 * nwav) {
        int r1 = r + nwav;                           // wave-uniform guard
        const float4* rp0 = (const float4*)(W + (size_t)r  * DCOLS);
        const float4* rp1 = (const float4*)(W + (size_t)r1 * DCOLS);

        float4 w0 = rp0[lane];                       // global_load_b128
        float4 w1 = make_float4(0.f, 0.f, 0.f, 0.f); // dummy row -> p=0 -> sq=0,
        if (r1 < C) w1 = rp1[lane];                  // s contribution 0 (self-mask)

        float p0 = w0.x*w0.x + w0.y*w0.y + w0.z*w0.z + w0.w*w0.w;
        float p1 = w1.x*w1.x + w1.y*w1.y + w1.z*w1.z + w1.w*w1.w;

        // two independent butterflies, interleaved by the scheduler
        p0 = SWZ_ADD(p0, 0x041F);  p1 = SWZ_ADD(p1, 0x041F);
        p0 = SWZ_ADD(p0, 0x081F);  p1 = SWZ_ADD(p1, 0x081F);
        p0 = SWZ_ADD(p0, 0x101F);  p1 = SWZ_ADD(p1, 0x101F);
        p0 = SWZ_ADD(p0, 0x201F);  p1 = SWZ_ADD(p1, 0x201F);
        p0 = SWZ_ADD(p0, 0x401F);  p1 = SWZ_ADD(p1, 0x401F);

        float inv0 = 1.0f / fmaxf(sqrtf(p0), EPSF); // F.normalize eps clamp
        float inv1 = 1.0f / fmaxf(sqrtf(p1), EPSF);
        ssq_acc += p0 * inv0 * inv0;                // sq_i of normalized rows
        ssq_acc += p1 * inv1 * inv1;                // == 0 for the dummy row

        s_acc.x += w0.x * inv0 + w1.x * inv1;
        s_acc.y += w0.y * inv0 + w1.y * inv1;
        s_acc.z += w0.z * inv0 + w1.z * inv1;
        s_acc.w += w0.w * inv0 + w1.w * inv1;
    }

    // per-block combine in LDS (ds_add_f32), then 128+1 global atomics/block
    atomicAdd(&s_block[4*lane + 0], s_acc.x);
    atomicAdd(&s_block[4*lane + 1], s_acc.y);
    atomicAdd(&s_block[4*lane + 2], s_acc.z);
    atomicAdd(&s_block[4*lane + 3], s_acc.w);
    if (lane == 0) atomicAdd(&ssq_block, ssq_acc);
    __syncthreads();
    if (tid < DCOLS) atomicAdd(&g_s[tid], s_block[tid]);
    if (tid == 0)    atomicAdd(g_ssq, ssq_block);
}

// ---------------------------------------------------------------------------
// K2: label gather (4096 rows, 2 MB -> resident in L2 after K1).
// 16 label rows per wave. Dots <w_norm[label], s> via V_WMMA_F32_16X16X4_F32:
//   A[m, 4c+k] = w_norm[label_m, 4c+k]   (ISA A-layout: lanes 0-15 hold K=0,1;
//                                         lanes 16-31 hold K=2,3 per chunk)
//   B[k, n]    = s[4c+k]  (broadcast over N)
// => D[m,n] = dot(row m, s) replicated over n; 32 chunks accumulate in fp32.
// ---------------------------------------------------------------------------
__global__ __launch_bounds__(256) void k2_gather_wmma(
        const float* __restrict__ W, const int* __restrict__ labels,
        int C, int B,
        const float* __restrict__ g_s, const float* __restrict__ g_ssq,
        float* __restrict__ g_total) {
    int tid   = threadIdx.x;
    int lane  = tid & 31;
    int wave  = tid >> 5;
    int gwave = blockIdx.x * (blockDim.x >> 5) + wave;
    int base  = gwave * 16;
    if (base >= B) return;                 // wave-uniform exit; EXEC stays full

    int m = lane & 15;                     // matrix row handled by this lane
    int h = lane >> 4;                     // K half: 0 -> K=0,1 ; 1 -> K=2,3
    int idx   = base + m;
    int valid = (idx < B);
    int cidx  = valid ? idx : (B - 1);
    int row   = labels[cidx];
    const float2* rp = (const float2*)(W + (size_t)row * DCOLS);

    // phase 1: row sumsq (each lane covers half the columns of its row)
    float part = 0.f;
    #pragma unroll 8
    for (int c = 0; c < 32; ++c) {
        float2 v = rp[2*c + h];            // cols 4c+2h, 4c+2h+1
        part += v.x*v.x + v.y*v.y;
    }
    float sumsq = SWZ_ADD(part, 0x401F);   // pair lanes m <-> m+16
    float inv = 1.0f / fmaxf(sqrtf(sumsq), EPSF);
    float sqi = sumsq * inv * inv;
    if (!valid) { inv = 0.f; sqi = 0.f; }  // select, EXEC stays full for WMMA

    // phase 2: WMMA dot accumulation (reloads hit L0/L2)
    v8f acc = {};
    #pragma unroll 4
    for (int c = 0; c < 32; ++c) {
        float2 v = rp[2*c + h];
        v2f a; a.x = v.x * inv;            a.y = v.y * inv;
        v2f b; b.x = g_s[4*c + 2*h];       b.y = g_s[4*c + 2*h + 1];
        acc = __builtin_amdgcn_wmma_f32_16x16x4_f32(
                  /*neg_a=*/false, a, /*neg_b=*/false, b,
                  /*c_mod=*/(short)0, acc, /*reuse_a=*/false, /*reuse_b=*/false);
    }
    // D[m,n]=dot_m for all n. Sum the 8 C/D VGPRs: lanes 0-15 -> sum_{m<8},
    // lanes 16-31 -> sum_{m>=8}; one xor-16 add -> full 16-row dot sum.
    float t = acc[0] + acc[1] + acc[2] + acc[3] + acc[4] + acc[5] + acc[6] + acc[7];
    float dots = t + __int_as_float(__builtin_amdgcn_ds_swizzle(__float_as_int(t), 0x401F));

    // sum sq_i over the 16 rows (each row duplicated in both halves -> *0.5)
    float q = wave_reduce_add(sqi) * 0.5f;

    if (lane == 0) {
        int nvalid = (B - base) < 16 ? (B - base) : 16;
        float contrib = (float)C * q + (float)nvalid * (*g_ssq) - 2.0f * dots;
        atomicAdd(g_total, contrib);
    }
}

// ---------------------------------------------------------------------------
// K3: finalize, matching reference op order: 1/total, /B, /(C-1)
// ---------------------------------------------------------------------------
__global__ void k3_finalize(const float* __restrict__ g_total,
                            float* __restrict__ out, int B, int C) {
    float loss = 1.0f / (*g_total);
    loss = loss / (float)B;
    loss = loss / (float)(C - 1);
    out[0] = loss;
}

// ---------------------------------------------------------------------------
extern "C" void kernel_launch(void* const* d_in, const int* in_sizes, int n_in,
                              void* d_out, int out_size, void* d_ws, size_t ws_size,
                              hipStream_t stream) {
    const float* W      = (const float*)d_in[0];
    const int*   labels = (const int*)d_in[1];
    int C = in_sizes[0] / DCOLS;   // 200000
    int B = in_sizes[1];           // 4096

    float* ws      = (float*)d_ws;
    float* g_s     = ws;           // [128]
    float* g_ssq   = ws + DCOLS;   // [1]
    float* g_total = ws + DCOLS+1; // [1]

    k0_zero<<<1, 256, 0, stream>>>(ws, DCOLS + 2);

    int blocksA = (C + 8 - 1) / 8;           // 8 waves/block
    if (blocksA > 1024) blocksA = 1024;      // ~8192 persistent waves, ~24 rows each
    k1_stream<<<blocksA, 256, 0, stream>>>(W, C, g_s, g_ssq);

    int wavesB  = (B + 15) / 16;
    int blocksB = (wavesB + 7) / 8;
    k2_gather_wmma<<<blocksB, 256, 0, stream>>>(W, labels, C, B, g_s, g_ssq, g_total);

    k3_finalize<<<1, 1, 0, stream>>>(g_total, (float*)d_out, B, C);
}